// AdultConnectome_28449863369169
// MI455X (gfx1250) — compile-verified
//
#include <hip/hip_runtime.h>

// A^6 @ x, COO -> CSR build once, then 6 atomic-free gather SpMMs.
// CDNA5 path: global_load_async_to_lds_b64 double-buffered edge streaming,
// s_wait_asynccnt / s_wait_dscnt pipeline control (wave32, per-wave ASYNCcnt).
// No WMMA by design: random-sparsity SpMM has no shared-B-operand structure
// (~0.08 nnz per 16x16 tile), so tensor ops cannot express the reduction.

#define SPMM_WAVES 8   // 8 waves * 32 lanes = 256 threads/block, 1 node per wave
#define CHUNK      64  // edges staged per pipeline step (2 x b64 async, 8B/edge)

__device__ __forceinline__ void async_load_b64(unsigned int lds_off, const void* gaddr) {
    // GV mode: 64-bit global address in VGPR pair, LDS byte offset in VGPR.
    asm volatile("global_load_async_to_lds_b64 %0, %1, off"
                 :: "v"(lds_off), "v"(gaddr)
                 : "memory");
}

#define ASYNC_WAIT_2() asm volatile("s_wait_asynccnt 2" ::: "memory")
#define ASYNC_WAIT_0() asm volatile("s_wait_asynccnt 0" ::: "memory")
#define DS_WAIT_0()    asm volatile("s_wait_dscnt 0"    ::: "memory")

__global__ void zero_kernel(unsigned int* p, int n) {
    int i = blockIdx.x * blockDim.x + threadIdx.x;
    if (i < n) p[i] = 0u;
}

__global__ void hist_kernel(const int* __restrict__ row, unsigned int* __restrict__ counts, int E) {
    int e = blockIdx.x * blockDim.x + threadIdx.x;
    if (e < E) atomicAdd(&counts[row[e]], 1u);
}

// Single-workgroup exclusive scan: counts[0..n) -> rowptr[0..n], rowptr[n] = total.
__global__ void scan_kernel(const unsigned int* __restrict__ counts,
                            unsigned int* __restrict__ rowptr, int n) {
    __shared__ unsigned int sums[1024];
    const int t  = threadIdx.x;
    const int ch = (n + 1023) / 1024;
    const int beg = t * ch;
    const int end = (beg + ch < n) ? (beg + ch) : n;
    unsigned int s = 0;
    for (int i = beg; i < end; ++i) s += counts[i];
    sums[t] = s;
    __syncthreads();
    for (int off = 1; off < 1024; off <<= 1) {
        unsigned int v = (t >= off) ? sums[t - off] : 0u;
        __syncthreads();
        sums[t] += v;
        __syncthreads();
    }
    unsigned int base = (t == 0) ? 0u : sums[t - 1];
    for (int i = beg; i < end; ++i) {
        rowptr[i] = base;
        base += counts[i];
    }
    if (t == 1023) rowptr[n] = sums[1023];
}

// Scatter edges into CSR order as interleaved 8B (col, val) pairs:
// low 32 bits = col index, high 32 bits = value bits.
__global__ void scatter_kernel(const int* __restrict__ row, const int* __restrict__ col,
                               const float* __restrict__ val,
                               const unsigned int* __restrict__ rowptr,
                               unsigned int* __restrict__ cursor,
                               unsigned long long* __restrict__ epair, int E) {
    int e = blockIdx.x * blockDim.x + threadIdx.x;
    if (e < E) {
        int r = row[e];
        unsigned int pos = rowptr[r] + atomicAdd(&cursor[r], 1u);
        unsigned int vb = __float_as_uint(val[e]);
        epair[pos] = ((unsigned long long)vb << 32) | (unsigned int)col[e];
    }
}

// Wave-per-node CSR SpMM. Edge (col,val) pairs stream double-buffered through
// LDS via async-to-LDS; lanes 0..23 hold the 24 features of the node's row.
__global__ __launch_bounds__(256) void spmm_kernel(
    const unsigned int* __restrict__ rowptr,
    const unsigned long long* __restrict__ epair,
    const float* __restrict__ xin,
    float* __restrict__ xout, int n)
{
    __shared__ unsigned long long s_edge[SPMM_WAVES][2][CHUNK];

    const int wave = threadIdx.x >> 5;
    const int lane = threadIdx.x & 31;
    const int node = blockIdx.x * SPMM_WAVES + wave;
    if (node >= n) return;

    const unsigned int beg = rowptr[node];
    const unsigned int end = rowptr[node + 1];
    const int deg = (int)(end - beg);

    // Lanes 24..31 compute a redundant feature (same address -> broadcast),
    // masked out at the store. Avoids per-iteration EXEC toggling and OOB.
    const int f = (lane < 24) ? lane : 23;
    float acc = 0.0f;

    if (deg > 0) {
        const unsigned int ebase =
            (unsigned int)(unsigned long long)&s_edge[wave][0][0];
        const unsigned int lds_lane = (unsigned int)lane * 8u;
        const int nch = (deg + CHUNK - 1) / CHUNK;

        // Prime chunk 0 into buffer 0: lanes cover edges [beg, beg+32) and
        // [beg+32, beg+64), one b64 (8B pair) each.
        DS_WAIT_0();
        async_load_b64(ebase + lds_lane,        epair + beg + lane);
        async_load_b64(ebase + 256u + lds_lane, epair + beg + 32 + lane);

        for (int c = 0; c < nch; ++c) {
            const int b = c & 1;
            if (c + 1 < nch) {
                const int nb = (c + 1) & 1;
                const unsigned int cbeg = beg + (unsigned int)(c + 1) * CHUNK;
                const unsigned int lbuf = ebase + (unsigned int)nb * (CHUNK * 8) + lds_lane;
                DS_WAIT_0();  // prior reads of buffer nb fully drained
                async_load_b64(lbuf,        epair + cbeg + lane);
                async_load_b64(lbuf + 256u, epair + cbeg + 32 + lane);
                ASYNC_WAIT_2();  // chunk c (issued earlier, in-order) is complete
            } else {
                ASYNC_WAIT_0();  // last chunk: drain everything
            }
            const int m = (deg - c * CHUNK < CHUNK) ? (deg - c * CHUNK) : CHUNK;
            #pragma unroll 8
            for (int i = 0; i < m; ++i) {
                const unsigned long long p = s_edge[wave][b][i];  // ds_load_b64 broadcast
                // Wave-uniform by construction -> force to SGPRs so the gather
                // becomes s_base + v_lane (GVS addressing) and FMA is vgpr*sgpr.
                const int   cc = __builtin_amdgcn_readfirstlane((int)(p & 0xffffffffull));
                const float v  = __uint_as_float(
                    (unsigned int)__builtin_amdgcn_readfirstlane((int)(p >> 32)));
                acc = fmaf(v, xin[cc * 24 + f], acc);  // coalesced 96B row gather
            }
        }
    }
    if (lane < 24) xout[node * 24 + lane] = acc;
}

extern "C" void kernel_launch(void* const* d_in, const int* in_sizes, int n_in,
                              void* d_out, int out_size, void* d_ws, size_t ws_size,
                              hipStream_t stream) {
    const float* x   = (const float*)d_in[0];
    const float* val = (const float*)d_in[1];
    const int*   row = (const int*)d_in[2];
    const int*   col = (const int*)d_in[3];
    const int E = in_sizes[1];
    const int N = in_sizes[0] / 24;

    // Workspace layout (256B-aligned): counts(+cursor), rowptr, epair(+pad),
    // ping, pong.  ~46 MB total.
    char* ws = (char*)d_ws;
    size_t off = 0;
    auto alloc = [&](size_t bytes) -> void* {
        void* p = ws + off;
        off += (bytes + 255) & ~(size_t)255;
        return p;
    };
    unsigned int*       counts = (unsigned int*)alloc((size_t)N * 4);   // reused as cursor
    unsigned int*       rowptr = (unsigned int*)alloc((size_t)(N + 1) * 4);
    unsigned long long* epair  = (unsigned long long*)alloc((size_t)E * 8 + 512); // pad: chunk overread
    float*              bufA   = (float*)alloc((size_t)N * 24 * 4);
    float*              bufB   = (float*)alloc((size_t)N * 24 * 4);

    const int B  = 256;
    const int gE = (E + B - 1) / B;
    const int gN = (N + B - 1) / B;

    // CSR build (once per call, deterministic work).
    zero_kernel<<<gN, B, 0, stream>>>(counts, N);
    hist_kernel<<<gE, B, 0, stream>>>(row, counts, E);
    scan_kernel<<<1, 1024, 0, stream>>>(counts, rowptr, N);
    zero_kernel<<<gN, B, 0, stream>>>(counts, N);  // counts -> cursor
    scatter_kernel<<<gE, B, 0, stream>>>(row, col, val, rowptr, counts, epair, E);

    // result = A^6 @ x  (ping-pong, final layer writes d_out directly)
    const int gs = (N + SPMM_WAVES - 1) / SPMM_WAVES;
    spmm_kernel<<<gs, B, 0, stream>>>(rowptr, epair, x,    bufA, N);
    spmm_kernel<<<gs, B, 0, stream>>>(rowptr, epair, bufA, bufB, N);
    spmm_kernel<<<gs, B, 0, stream>>>(rowptr, epair, bufB, bufA, N);
    spmm_kernel<<<gs, B, 0, stream>>>(rowptr, epair, bufA, bufB, N);
    spmm_kernel<<<gs, B, 0, stream>>>(rowptr, epair, bufB, bufA, N);
    spmm_kernel<<<gs, B, 0, stream>>>(rowptr, epair, bufA, (float*)d_out, N);
}